// GateRecurrent2dnoind_23081154249325
// MI455X (gfx1250) — compile-verified
//
#include <hip/hip_runtime.h>
#include <stdint.h>

// GateRecurrent2d: H[h,w] = B*X + G1*Hprev[h-1] + G2*Hprev[h] + G3*Hprev[h+1]
// Memory-bound streaming scan (403 MB moved for 117 MFLOP => ~17us floor at
// 23.3 TB/s). CDNA5 path used: async global->LDS staging (ASYNCcnt) to
// double-buffer the 5 input streams around the serial w-dependency chain.

#define HDIM 256
#define WDIM 256
#define TW   4                 // tile width along w (one b128 per array/row)
#define NTILES (WDIM / TW)     // 64

// One global_load_async_to_lds_b128: 16B global (row-contiguous along w)
// -> 16B LDS. GVS addressing: 64-bit SGPR base + 32-bit per-lane offset.
__device__ __forceinline__ void async_row_b128(uint32_t lds_addr, uint32_t voff,
                                               uint64_t sbase) {
  asm volatile("global_load_async_to_lds_b128 %0, %1, %2"
               :: "v"(lds_addr), "v"(voff), "s"(sbase)
               : "memory");
}

__global__ __launch_bounds__(HDIM, 1)
void gate_recurrent_2d_kernel(const float* __restrict__ X,
                              const float* __restrict__ B,
                              const float* __restrict__ G1,
                              const float* __restrict__ G2,
                              const float* __restrict__ G3,
                              float* __restrict__ Out) {
  // Double-buffered input stage: 2 x 5 arrays x 256 rows x TW floats = 40 KB.
  __shared__ float tile[2][5][HDIM][TW];
  // Ping-pong recurrence columns with zero borders (implements the pads).
  __shared__ float hcol[2][HDIM + 2];

  const int tid = threadIdx.x;                       // row h
  const long long planeOff = (long long)blockIdx.x * (HDIM * WDIM);

  const uint64_t xa  = (uint64_t)(uintptr_t)(X  + planeOff);
  const uint64_t ba  = (uint64_t)(uintptr_t)(B  + planeOff);
  const uint64_t g1a = (uint64_t)(uintptr_t)(G1 + planeOff);
  const uint64_t g2a = (uint64_t)(uintptr_t)(G2 + planeOff);
  const uint64_t g3a = (uint64_t)(uintptr_t)(G3 + planeOff);
  float* ob = Out + planeOff + (long long)tid * WDIM;

  // Zero both H columns including the two border cells (never written again).
  hcol[0][tid + 1] = 0.0f;
  hcol[1][tid + 1] = 0.0f;
  if (tid < 2) {
    hcol[0][tid * (HDIM + 1)] = 0.0f;   // entries 0 and 257
    hcol[1][tid * (HDIM + 1)] = 0.0f;
  }

  // Precompute this thread's LDS destination byte addresses (AS(3) offsets).
  uint32_t ldsa[2][5];
#pragma unroll
  for (int bf = 0; bf < 2; ++bf)
#pragma unroll
    for (int a = 0; a < 5; ++a)
      ldsa[bf][a] = (uint32_t)(uintptr_t)&tile[bf][a][tid][0];

  const uint32_t rowByte = (uint32_t)tid * (WDIM * 4);

  // Prime the pipeline: stage tile 0 asynchronously.
  {
    const uint32_t voff = rowByte;  // w0 = 0
    async_row_b128(ldsa[0][0], voff, xa);
    async_row_b128(ldsa[0][1], voff, ba);
    async_row_b128(ldsa[0][2], voff, g1a);
    async_row_b128(ldsa[0][3], voff, g2a);
    async_row_b128(ldsa[0][4], voff, g3a);
  }
  __syncthreads();  // hcol zero-init visible to neighbors

  float hprev = 0.0f;  // Hprev[tid] lives in a register

  for (int t = 0; t < NTILES; ++t) {
    const int buf = t & 1;

    if (t + 1 < NTILES) {
      // Prefetch next tile into the other buffer, then wait for current tile:
      // <=5 outstanding per wave means the older 5 (this buffer) completed.
      const uint32_t voff = rowByte + (uint32_t)(t + 1) * (TW * 4);
      async_row_b128(ldsa[buf ^ 1][0], voff, xa);
      async_row_b128(ldsa[buf ^ 1][1], voff, ba);
      async_row_b128(ldsa[buf ^ 1][2], voff, g1a);
      async_row_b128(ldsa[buf ^ 1][3], voff, g2a);
      async_row_b128(ldsa[buf ^ 1][4], voff, g3a);
      asm volatile("s_wait_asynccnt 5" ::: "memory");
    } else {
      asm volatile("s_wait_asynccnt 0" ::: "memory");
    }

    // Pull this row's staged data into registers (one ds_load_b128 each).
    float xr[TW], br[TW], g1r[TW], g2r[TW], g3r[TW], outr[TW];
    *(float4*)xr  = *(const float4*)&tile[buf][0][tid][0];
    *(float4*)br  = *(const float4*)&tile[buf][1][tid][0];
    *(float4*)g1r = *(const float4*)&tile[buf][2][tid][0];
    *(float4*)g2r = *(const float4*)&tile[buf][3][tid][0];
    *(float4*)g3r = *(const float4*)&tile[buf][4][tid][0];

#pragma unroll
    for (int j = 0; j < TW; ++j) {
      // TW is even, so parity of absolute w == parity of j.
      const float* rb = hcol[j & 1];         // holds H(:, w-1)
      float*       wb = hcol[(j & 1) ^ 1];   // receives H(:, w)
      const float up = rb[tid];              // Hprev[h-1] (0 at h==0)
      const float dn = rb[tid + 2];          // Hprev[h+1] (0 at h==255)
      const float h  = br[j] * xr[j] + g1r[j] * up + g2r[j] * hprev + g3r[j] * dn;
      wb[tid + 1] = h;
      hprev  = h;
      outr[j] = h;
      __syncthreads();  // one barrier per w step: write(w) before read(w+1)
    }

    // Coalesce over time: one b128 store per tile; each 128B line is fully
    // written by this thread across 8 consecutive tiles (L2 wr-rinse merges).
    *(float4*)&ob[t * TW] = *(float4*)outr;
  }
}

extern "C" void kernel_launch(void* const* d_in, const int* in_sizes, int n_in,
                              void* d_out, int out_size, void* d_ws, size_t ws_size,
                              hipStream_t stream) {
  const float* X  = (const float*)d_in[0];
  const float* B  = (const float*)d_in[1];
  const float* G1 = (const float*)d_in[2];
  const float* G2 = (const float*)d_in[3];
  const float* G3 = (const float*)d_in[4];
  float* Out = (float*)d_out;

  const int planes = in_sizes[0] / (HDIM * WDIM);  // N*C = 256
  gate_recurrent_2d_kernel<<<dim3(planes), dim3(HDIM), 0, stream>>>(
      X, B, G1, G2, G3, Out);

  (void)n_in; (void)out_size; (void)d_ws; (void)ws_size;
}